// GaussianMixture_41626823033345
// MI455X (gfx1250) — compile-verified
//
#include <hip/hip_runtime.h>
#include <hip/hip_bf16.h>

// GMM log-likelihood: N=65536 samples (D=2), M=1024 components.
// Score matrix S (M x N) = Coef(M x 8) * Feat(8 x N)  -> two chained
// V_WMMA_F32_16X16X4_F32 per 16x16 tile (K = 8), then online logsumexp
// over the component (row) dimension, which lives in-lane in the D layout.

typedef __attribute__((ext_vector_type(2))) float v2f;
typedef __attribute__((ext_vector_type(8))) float v8f;

#define GM_N 65536
#define GM_M 1024

// ---------------------------------------------------------------------------
// Kernel 1: per-component coefficient build (single block of 1024 threads).
//   gamma = A A^T (2x2 symmetric), det, log_softmax(w), fold everything into
//   coef[j][0..7] = { -g00, -2*g01, -g11, +2*gm0, +2*gm1, logw - m'Gm, 0, 0 }
// ---------------------------------------------------------------------------
__global__ __launch_bounds__(GM_M) void gm_coef_kernel(
    const float* __restrict__ mu,    // (M,2)
    const float* __restrict__ Amat,  // (M,2,2)
    const float* __restrict__ w,     // (M,1)
    float* __restrict__ coef)        // (M,8) out
{
    __shared__ float red[GM_M];
    const int j = threadIdx.x;

    const float wj = w[j];

    // logsumexp(w) over M: max reduce, then sum-exp reduce.
    red[j] = wj;
    __syncthreads();
    #pragma unroll
    for (int s = GM_M / 2; s > 0; s >>= 1) {
        if (j < s) red[j] = fmaxf(red[j], red[j + s]);
        __syncthreads();
    }
    const float wmax = red[0];
    __syncthreads();
    red[j] = __expf(wj - wmax);
    __syncthreads();
    #pragma unroll
    for (int s = GM_M / 2; s > 0; s >>= 1) {
        if (j < s) red[j] += red[j + s];
        __syncthreads();
    }
    const float lse_w = wmax + __logf(red[0]);

    // gamma = A A^T
    const float a00 = Amat[j * 4 + 0], a01 = Amat[j * 4 + 1];
    const float a10 = Amat[j * 4 + 2], a11 = Amat[j * 4 + 3];
    const float g00 = a00 * a00 + a01 * a01;
    const float g01 = a00 * a10 + a01 * a11;   // == g10
    const float g11 = a10 * a10 + a11 * a11;
    const float det = g00 * g11 - g01 * g01;

    const float logw = (wj - lse_w) + 0.5f * __logf(det);

    const float m0 = mu[j * 2 + 0], m1 = mu[j * 2 + 1];
    const float gm0 = g00 * m0 + g01 * m1;
    const float gm1 = g01 * m0 + g11 * m1;
    const float mGm = gm0 * m0 + gm1 * m1;

    float* c = coef + j * 8;
    c[0] = -g00;
    c[1] = -2.0f * g01;
    c[2] = -g11;
    c[3] = 2.0f * gm0;
    c[4] = 2.0f * gm1;
    c[5] = logw - mGm;
    c[6] = 0.0f;
    c[7] = 0.0f;
}

// ---------------------------------------------------------------------------
// Kernel 2: main WMMA + streaming logsumexp.
// Block = 256 threads (8 wave32), each wave owns 16 sample columns.
// Loop over 64 component tiles of 16; per tile: two f32 WMMAs (K=4 each),
// then per-lane tile-max + rescaled accumulation (10 exps / 8 values).
// ---------------------------------------------------------------------------
__global__ __launch_bounds__(256) void gm_main_kernel(
    const float* __restrict__ sample,  // (N,2)
    const float* __restrict__ coef,    // (M,8)
    float* __restrict__ out)           // (N)
{
    __shared__ float lcoef[GM_M * 8];  // 32 KB

    // Cooperative stage of all component coefficients into LDS (float4).
    {
        const float4* src = (const float4*)coef;
        float4* dst = (float4*)lcoef;
        #pragma unroll
        for (int i = threadIdx.x; i < GM_M * 8 / 4; i += 256) dst[i] = src[i];
    }
    __syncthreads();

    const int lane  = threadIdx.x & 31;
    const int wave  = threadIdx.x >> 5;
    const int half  = lane >> 4;    // 0: lanes 0-15, 1: lanes 16-31
    const int ccol  = lane & 15;    // column within 16-wide tile

    const int rowBase = blockIdx.x * 128 + wave * 16;  // first sample index
    const int sIdx = rowBase + ccol;

    // Features for this lane's sample column: [x0^2, x0x1, x1^2, x0, x1, 1, 0, 0]
    const float x0 = sample[2 * sIdx + 0];
    const float x1 = sample[2 * sIdx + 1];
    const float f0 = x0 * x0, f1 = x0 * x1, f2 = x1 * x1;
    const float f3 = x0, f4 = x1, f5 = 1.0f, f6 = 0.0f, f7 = 0.0f;

    // B-matrix fragments (4x16 f32, 2 VGPRs): VGPR v holds K = v + 2*half.
    v2f bLo, bHi;
    bLo.x = half ? f2 : f0;  bLo.y = half ? f3 : f1;   // K = {0,1} | {2,3}
    bHi.x = half ? f6 : f4;  bHi.y = half ? f7 : f5;   // K = {4,5} | {6,7}

    float mx  = -__builtin_inff();
    float sum = 0.0f;

    #pragma unroll 4
    for (int t = 0; t < GM_M / 16; ++t) {
        // A-matrix fragment (16x4 f32): lane holds component row (lane&15),
        // VGPR v holds K = v + 2*half.
        const float* cr = &lcoef[(t * 16 + ccol) * 8];
        v2f aLo, aHi;
        aLo.x = cr[2 * half + 0];  aLo.y = cr[2 * half + 1];
        aHi.x = cr[2 * half + 4];  aHi.y = cr[2 * half + 5];

        v8f acc = {0.f, 0.f, 0.f, 0.f, 0.f, 0.f, 0.f, 0.f};
        acc = __builtin_amdgcn_wmma_f32_16x16x4_f32(
                  false, aLo, false, bLo, (short)0, acc, false, false);
        acc = __builtin_amdgcn_wmma_f32_16x16x4_f32(
                  false, aHi, false, bHi, (short)0, acc, false, false);

        // acc[k] = s[(t*16 + k + 8*half) , sample col]  (8 component rows/lane)
        float tm = acc[0];
        #pragma unroll
        for (int k = 1; k < 8; ++k) tm = fmaxf(tm, acc[k]);
        float ts = 0.0f;
        #pragma unroll
        for (int k = 0; k < 8; ++k) ts += __expf(acc[k] - tm);

        const float m2 = fmaxf(mx, tm);
        sum = sum * __expf(mx - m2) + ts * __expf(tm - m2);
        mx  = m2;
    }

    // Merge the two half-wave partials (other 8 component rows per tile).
    const float mxo  = __shfl_xor(mx, 16, 32);
    const float sumo = __shfl_xor(sum, 16, 32);
    const float m2 = fmaxf(mx, mxo);
    const float res = m2 + __logf(sum * __expf(mx - m2) + sumo * __expf(mxo - m2));

    if (lane < 16) out[rowBase + lane] = res;
}

// ---------------------------------------------------------------------------
extern "C" void kernel_launch(void* const* d_in, const int* in_sizes, int n_in,
                              void* d_out, int out_size, void* d_ws, size_t ws_size,
                              hipStream_t stream) {
    const float* sample = (const float*)d_in[0];  // (N,2)
    const float* mu     = (const float*)d_in[1];  // (M,2)
    const float* Amat   = (const float*)d_in[2];  // (M,2,2)
    const float* w      = (const float*)d_in[3];  // (M,1)
    float* out = (float*)d_out;

    float* coef = (float*)d_ws;  // M*8 floats = 32 KB scratch

    gm_coef_kernel<<<1, GM_M, 0, stream>>>(mu, Amat, w, coef);

    const int n = in_sizes[0] / 2;          // N samples
    const int blocks = n / 128;             // 16 samples/wave * 8 waves/block
    gm_main_kernel<<<blocks, 256, 0, stream>>>(sample, coef, out);
}